// LightGCN_42193758716127
// MI455X (gfx1250) — compile-verified
//
#include <hip/hip_runtime.h>
#include <hip/hip_bf16.h>

#define USER_CNT   100000
#define ITEM_CNT   50000
#define NTOT       (USER_CNT + ITEM_CNT)   // 150000
#define EMB        64
#define NUM_LAYERS 3
#define NNZ        4800000
#define BATCH      16384
#define NCHUNKS    (NNZ / 32)              // 150000 (NNZ divisible by 32)

typedef __attribute__((ext_vector_type(2))) float v2f;
typedef __attribute__((ext_vector_type(8))) float v8f;

// ---------------------------------------------------------------------------
// acc = x = concat(user_emb, item_emb), float4-vectorized
// ---------------------------------------------------------------------------
__global__ void lgcn_init(const float* __restrict__ ue, const float* __restrict__ ie,
                          float* __restrict__ acc, float* __restrict__ x) {
    size_t i = (size_t)blockIdx.x * blockDim.x + threadIdx.x;       // float4 units
    const size_t tot4  = (size_t)NTOT * EMB / 4;
    const size_t ucnt4 = (size_t)USER_CNT * EMB / 4;
    if (i >= tot4) return;
    float4 v = (i < ucnt4) ? ((const float4*)ue)[i] : ((const float4*)ie)[i - ucnt4];
    ((float4*)acc)[i] = v;
    ((float4*)x)[i]   = v;
}

__global__ void lgcn_zero(float* __restrict__ p) {
    size_t i = (size_t)blockIdx.x * blockDim.x + threadIdx.x;
    if (i < (size_t)NTOT * EMB / 4) ((float4*)p)[i] = make_float4(0.f, 0.f, 0.f, 0.f);
}

__global__ void lgcn_addacc(float* __restrict__ acc, const float* __restrict__ x) {
    size_t i = (size_t)blockIdx.x * blockDim.x + threadIdx.x;
    if (i >= (size_t)NTOT * EMB / 4) return;
    float4 a = ((const float4*)acc)[i];
    float4 b = ((const float4*)x)[i];
    ((float4*)acc)[i] = make_float4(a.x + b.x, a.y + b.y, a.z + b.z, a.w + b.w);
}

// ---------------------------------------------------------------------------
// COO SpMM scatter: xout[r] += val * xin[c].
// One wave32 owns chunks of 32 edges (grid-stride). Edge lists are streamed
// with non-temporal loads (keep the 38.4MB L2-resident state hot) and the
// next chunk is prefetched (global_prefetch_b8). Each edge is broadcast via
// __shfl; each lane handles 2 of the 64 embedding floats -> coalesced float2
// gather + 2 coalesced global_atomic_add_f32 (no-return, STOREcnt path).
// ---------------------------------------------------------------------------
__global__ void lgcn_spmm(const int* __restrict__ rows, const int* __restrict__ cols,
                          const float* __restrict__ vals,
                          const float* __restrict__ xin, float* __restrict__ xout) {
    const int lane   = threadIdx.x & 31;
    const int wave   = (int)((blockIdx.x * blockDim.x + threadIdx.x) >> 5);
    const int nwaves = (int)((gridDim.x * blockDim.x) >> 5);

    for (int chunk = wave; chunk < NCHUNKS; chunk += nwaves) {
        const int e0  = chunk * 32;
        const int nxt = chunk + nwaves;
        if (nxt < NCHUNKS) {                 // wave-uniform
            __builtin_prefetch(rows + nxt * 32 + lane, 0, 0);
            __builtin_prefetch(cols + nxt * 32 + lane, 0, 0);
            __builtin_prefetch(vals + nxt * 32 + lane, 0, 0);
        }
        int   r = __builtin_nontemporal_load(rows + e0 + lane);
        int   c = __builtin_nontemporal_load(cols + e0 + lane);
        float v = __builtin_nontemporal_load(vals + e0 + lane);

#pragma unroll 4
        for (int i = 0; i < 32; ++i) {
            const int   ri = __shfl(r, i, 32);
            const int   ci = __shfl(c, i, 32);
            const float vi = __shfl(v, i, 32);
            const float2 xv = *(const float2*)(xin + (size_t)ci * EMB + lane * 2);
            atomicAdd(xout + (size_t)ri * EMB + lane * 2 + 0, vi * xv.x);
            atomicAdd(xout + (size_t)ri * EMB + lane * 2 + 1, vi * xv.y);
        }
    }
}

// ---------------------------------------------------------------------------
// Final scoring with fp32 WMMA: one wave computes a 16x16 tile of U·I^T via
// 16 chained v_wmma_f32_16x16x4_f32 over K=64 and extracts the diagonal.
// f32 A(16x4)/B(4x16) fragment layout (ISA 7.12.2): elem = lane%16,
//   k = vgpr + 2*(lane/16)  -> per k-step each lane loads one float2 of its
// user row (A) and one float2 of its item row (B).
// C/D layout: lane L holds N = L%16, M = comp + 8*(L/16); the diagonal sits
// in lanes 0..7 (comp = lane) and lanes 24..31 (comp = lane-24).
// final = acc/4 for both operands -> scale tile by 1/16.
// ---------------------------------------------------------------------------
__global__ void lgcn_dot_wmma(const int* __restrict__ uidx, const int* __restrict__ iidx,
                              const float* __restrict__ acc, float* __restrict__ out) {
    const int lane = threadIdx.x & 31;
    const int tile = (int)(blockIdx.x * (blockDim.x >> 5) + (threadIdx.x >> 5));
    if (tile * 16 >= BATCH) return;          // wave-uniform (grid is exact)

    const int m = lane & 15;                 // pair index within tile
    const int h = lane >> 4;                 // lane half selects k phase
    const int p = tile * 16 + m;

    const int urow = uidx[p];
    const int irow = USER_CNT + iidx[p];
    const float* ub = acc + (size_t)urow * EMB + 2 * h;
    const float* ib = acc + (size_t)irow * EMB + 2 * h;

    v8f cacc = {0.f, 0.f, 0.f, 0.f, 0.f, 0.f, 0.f, 0.f};
#pragma unroll
    for (int k0 = 0; k0 < EMB; k0 += 4) {
        v2f a = *(const v2f*)(ub + k0);      // A[m][k0 + 2h + {0,1}]
        v2f b = *(const v2f*)(ib + k0);      // B[k0 + 2h + {0,1}][m] == I[m][..]
        cacc = __builtin_amdgcn_wmma_f32_16x16x4_f32(
            /*neg_a=*/false, a, /*neg_b=*/false, b,
            /*c_mod=*/(short)0, cacc, /*reuse_a=*/false, /*reuse_b=*/false);
    }

    const int comp = m - 8 * h;              // diagonal component for this lane
    if (comp >= 0 && comp < 8) {
        out[p] = cacc[comp] * 0.0625f;       // (acc/4)·(acc/4)
    }
}

// ---------------------------------------------------------------------------
extern "C" void kernel_launch(void* const* d_in, const int* in_sizes, int n_in,
                              void* d_out, int out_size, void* d_ws, size_t ws_size,
                              hipStream_t stream) {
    (void)in_sizes; (void)n_in; (void)out_size; (void)ws_size;
    const int*   uidx = (const int*)  d_in[0];
    const int*   iidx = (const int*)  d_in[1];
    const float* ue   = (const float*)d_in[2];
    const float* ie   = (const float*)d_in[3];
    const int*   rows = (const int*)  d_in[4];
    const int*   cols = (const int*)  d_in[5];
    const float* vals = (const float*)d_in[6];
    float*       out  = (float*)      d_out;

    const size_t NE = (size_t)NTOT * EMB;    // 9.6M floats
    float* acc = (float*)d_ws;
    float* xa  = acc + NE;
    float* xb  = xa + NE;

    const int vec4   = (int)(NE / 4);                    // 2.4M float4
    const int vblk   = (vec4 + 255) / 256;
    lgcn_init<<<vblk, 256, 0, stream>>>(ue, ie, acc, xa);

    float* xin = xa;
    float* xout = xb;
    for (int l = 0; l < NUM_LAYERS; ++l) {
        lgcn_zero<<<vblk, 256, 0, stream>>>(xout);
        lgcn_spmm<<<2048, 256, 0, stream>>>(rows, cols, vals, xin, xout);
        lgcn_addacc<<<vblk, 256, 0, stream>>>(acc, xout);
        float* t = xin; xin = xout; xout = t;
    }

    // 1024 tiles of 16 pairs; 8 waves/block -> 128 blocks exactly
    lgcn_dot_wmma<<<BATCH / 16 / 8, 256, 0, stream>>>(uidx, iidx, acc, out);
}